// Network_31688268710403
// MI455X (gfx1250) — compile-verified
//
#include <hip/hip_runtime.h>
#include <math.h>

#define NP        262144
#define POSE_NUM  128
#define NCOMP     48
#define GRID_SZ   512
#define POSE_DIM  20
#define PADC      32          // feat columns padded to 32 in LDS (cols 20..31 = 0)
#define NJ        4
#define WAVES_PER_BLOCK 8

typedef float v2f __attribute__((ext_vector_type(2)));
typedef float v8f __attribute__((ext_vector_type(8)));

// ---------------------------------------------------------------------------
// Kernel 1: quaternion-similarity top-1 over 128 key poses -> ws[0]=w, ws[1]=sel
// ---------------------------------------------------------------------------
__device__ __forceinline__ void aa2quat(float ax, float ay, float az, float q[4]) {
  float angle = sqrtf(ax * ax + ay * ay + az * az);
  float half  = 0.5f * angle;
  bool  sm    = fabsf(angle) < 1e-6f;
  float safe  = sm ? 1.0f : angle;
  float sfac  = sm ? (0.5f - angle * angle * (1.0f / 48.0f)) : (sinf(half) / safe);
  q[0] = cosf(half);
  q[1] = ax * sfac;
  q[2] = ay * sfac;
  q[3] = az * sfac;
}

__global__ void __launch_bounds__(POSE_NUM) topk_kernel(
    const float* __restrict__ poses,        // (1,72)
    const float* __restrict__ key_poses,    // (128,72)
    const int*  __restrict__ part_indices,  // (4,)
    float* __restrict__ ws) {
  __shared__ float dist_s[POSE_NUM];
  int p = threadIdx.x;
  float d = 0.0f;
  for (int j = 0; j < NJ; ++j) {
    int joint = part_indices[j];
    float qq[4], kq[4];
    aa2quat(poses[joint * 3 + 0], poses[joint * 3 + 1], poses[joint * 3 + 2], qq);
    aa2quat(key_poses[p * 72 + joint * 3 + 0],
            key_poses[p * 72 + joint * 3 + 1],
            key_poses[p * 72 + joint * 3 + 2], kq);
    d += fabsf(qq[0] * kq[0] + qq[1] * kq[1] + qq[2] * kq[2] + qq[3] * kq[3]);
  }
  dist_s[p] = d;
  __syncthreads();
  if (p == 0) {
    float best = dist_s[0];
    int   bi   = 0;
    for (int i = 1; i < POSE_NUM; ++i)
      if (dist_s[i] > best) { best = dist_s[i]; bi = i; }   // first max, like top_k
    float wgt = best / fmaxf(fabsf(best), 1e-16f);          // L1 normalize (k=1)
    ws[0] = wgt;
    ((int*)ws)[1] = bi;
  }
}

// ---------------------------------------------------------------------------
// Kernel 2: fused sample + triple-plane GEMM via V_WMMA_F32_16X16X4_F32.
// One wave handles a 16-point M-tile; K=48 -> 12 WMMA steps; N=20 -> 2 tiles
// (second tile columns 20..31 are zero-padded in LDS so all B loads are
//  unconditional ds_loads -- no exec-mask toggling in the WMMA stream).
// ---------------------------------------------------------------------------
__global__ void __launch_bounds__(32 * WAVES_PER_BLOCK) fuse_kernel(
    const float* __restrict__ pts,         // (NP,3)
    const float* __restrict__ tbounds,     // (1,2,3)
    const float* __restrict__ coord_line,  // (3,128,48,512,1)
    const float* __restrict__ feat_line,   // (3,128,48,20,1)
    const float* __restrict__ ws,
    float* __restrict__ out) {             // (1,NP,60)
  __shared__ float featLds[3 * NCOMP * PADC];  // 4608 f32 = 18.4 KB

  const float w   = ws[0];
  const int   sel = ((const int*)ws)[1];

  // Stage feat_line[:, sel] into LDS, padded to 32 columns (pad = 0).
  for (int i = threadIdx.x; i < 3 * NCOMP * PADC; i += blockDim.x) {
    int col = i & (PADC - 1);
    int row = i >> 5;                      // plane*NCOMP + c
    int pl  = row / NCOMP;
    int c   = row - pl * NCOMP;
    float v = 0.0f;
    if (col < POSE_DIM)
      v = feat_line[((pl * POSE_NUM + sel) * NCOMP + c) * POSE_DIM + col];
    featLds[i] = v;
  }
  __syncthreads();

  const int lane   = threadIdx.x & 31;
  const int waveId = threadIdx.x >> 5;
  const int half   = lane >> 4;   // 0: K%4 in {0,1}   1: K%4 in {2,3}
  const int m      = lane & 15;   // point within tile / column within N-tile
  const int base   = (blockIdx.x * WAVES_PER_BLOCK + waveId) * 16;
  const int pt     = base + m;

  // --- normalize point, per-axis sample position (bilinear, align_corners) ---
  int   i0[3], i1[3];
  float fr[3];
#pragma unroll
  for (int axis = 0; axis < 3; ++axis) {
    float lo  = tbounds[axis];
    float hi  = tbounds[3 + axis];
    float x   = pts[pt * 3 + axis];
    float nx  = (x - lo) / (hi - lo) * 2.0f - 1.0f;
    float pos = (nx + 1.0f) * 0.5f * (float)(GRID_SZ - 1);
    pos = fminf(fmaxf(pos, 0.0f), (float)(GRID_SZ - 1));
    int ii0 = (int)floorf(pos);
    int ii1 = ii0 + 1;
    if (ii1 > GRID_SZ - 1) ii1 = GRID_SZ - 1;
    i0[axis] = ii0;
    i1[axis] = ii1;
    fr[axis] = pos - (float)ii0;
  }

  // --- gather this lane's 24 components per axis (K = 4t + 2*half + e) ------
  // Matches the 16x4 f32 A-fragment layout: lanes0-15 hold K%4 in {0,1},
  // lanes16-31 hold K%4 in {2,3}.  All reads hit L2 (table is 294 KB).
  float crd[3][12][2];
#pragma unroll
  for (int axis = 0; axis < 3; ++axis) {
    const float* tbl = coord_line + (axis * POSE_NUM + sel) * (NCOMP * GRID_SZ);
    float f = fr[axis], omf = 1.0f - fr[axis];
#pragma unroll
    for (int t = 0; t < 12; ++t) {
#pragma unroll
      for (int e = 0; e < 2; ++e) {
        int c = 4 * t + 2 * half + e;
        const float* row = tbl + c * GRID_SZ;
        crd[axis][t][e] = row[i0[axis]] * omf + row[i1[axis]] * f;
      }
    }
  }

  // --- three plane GEMMs on the matrix pipe ---------------------------------
  const int axA[3] = {0, 0, 1};
  const int axB[3] = {1, 2, 2};
  const int n1     = 16 + m;                 // second-N-tile column (padded)

#pragma unroll
  for (int p = 0; p < 3; ++p) {
    v8f acc0 = {};
    v8f acc1 = {};
    const float* fb = &featLds[p * NCOMP * PADC];
#pragma unroll
    for (int t = 0; t < 12; ++t) {
      const int c0 = 4 * t + 2 * half;
      v2f A, B0, B1;
      A.x  = crd[axA[p]][t][0] * crd[axB[p]][t][0];
      A.y  = crd[axA[p]][t][1] * crd[axB[p]][t][1];
      B0.x = fb[c0 * PADC + m];
      B0.y = fb[(c0 + 1) * PADC + m];
      B1.x = fb[c0 * PADC + n1];
      B1.y = fb[(c0 + 1) * PADC + n1];
      acc0 = __builtin_amdgcn_wmma_f32_16x16x4_f32(false, A, false, B0,
                                                   (short)0, acc0, false, false);
      acc1 = __builtin_amdgcn_wmma_f32_16x16x4_f32(false, A, false, B1,
                                                   (short)0, acc1, false, false);
    }
    // D layout: lane = N column, VGPR r = M row r (lanes0-15) / 8+r (lanes16-31)
#pragma unroll
    for (int r = 0; r < 8; ++r) {
      int mm = r + 8 * half;
      int o  = (base + mm) * 60 + p * 20;
      out[o + m] = w * acc0[r];
      if (m < POSE_DIM - 16) out[o + 16 + m] = w * acc1[r];
    }
  }
}

// ---------------------------------------------------------------------------
extern "C" void kernel_launch(void* const* d_in, const int* in_sizes, int n_in,
                              void* d_out, int out_size, void* d_ws, size_t ws_size,
                              hipStream_t stream) {
  const float* pts          = (const float*)d_in[0];
  const float* poses        = (const float*)d_in[1];
  const float* key_poses    = (const float*)d_in[2];
  const float* tbounds      = (const float*)d_in[3];
  const float* coord_line   = (const float*)d_in[4];
  const float* feat_line    = (const float*)d_in[5];
  const int*   part_indices = (const int*)d_in[6];
  float*       out          = (float*)d_out;
  float*       ws           = (float*)d_ws;

  topk_kernel<<<1, POSE_NUM, 0, stream>>>(poses, key_poses, part_indices, ws);

  const int blocks = NP / (WAVES_PER_BLOCK * 16);  // 2048
  fuse_kernel<<<blocks, 32 * WAVES_PER_BLOCK, 0, stream>>>(
      pts, tbounds, coord_line, feat_line, ws, out);
}